// SAGE_12043088298174
// MI455X (gfx1250) — compile-verified
//
#include <hip/hip_runtime.h>

typedef __attribute__((ext_vector_type(2))) float v2f;
typedef __attribute__((ext_vector_type(8))) float v8f;

// ---------------------------------------------------------------------------
// Degree / inverse-degree
// ---------------------------------------------------------------------------
__global__ void sage_deg_kernel(const int* __restrict__ dst, float* __restrict__ deg, int nE) {
    int t = blockIdx.x * blockDim.x + threadIdx.x;
    if (t < nE) atomicAdd(&deg[dst[t]], 1.0f);
}

__global__ void sage_invdeg_kernel(const float* __restrict__ deg, float* __restrict__ invdeg, int n) {
    int t = blockIdx.x * blockDim.x + threadIdx.x;
    if (t < n) invdeg[t] = 1.0f / fmaxf(deg[t], 1.0f);
}

// ---------------------------------------------------------------------------
// Edge scatter-add: agg[dst] += h[src]   (DIM floats per node)
// One edge handled by DIM/4 threads; float4 load, 4 f32 atomics per thread.
// Feature matrices (51.2MB) are L2-resident on MI455X (192MB L2), so the
// random gather and the atomic scatter both resolve in L2.
// ---------------------------------------------------------------------------
template <int DIM>
__global__ void sage_scatter_kernel(const float* __restrict__ H,
                                    const int* __restrict__ src,
                                    const int* __restrict__ dst,
                                    float* __restrict__ Agg, int nE) {
    constexpr int LPE = DIM / 4;  // lanes per edge
    long long t = (long long)blockIdx.x * blockDim.x + threadIdx.x;
    int e = (int)(t / LPE);
    if (e >= nE) return;
    int c = (int)(t % LPE) * 4;
    int s = src[e];
    int d = dst[e];
    const float4 v = *(const float4*)(H + (size_t)s * DIM + c);
    float* o = Agg + (size_t)d * DIM + c;
    atomicAdd(o + 0, v.x);
    atomicAdd(o + 1, v.y);
    atomicAdd(o + 2, v.z);
    atomicAdd(o + 3, v.w);
}

// ---------------------------------------------------------------------------
// Fused SAGE layer GEMM using V_WMMA_F32_16X16X4_F32 (full fp32 precision):
//   Out[16-row strip] = Hself @ Ws + (Agg * invdeg[row]) @ Wn + bias  (opt ReLU)
// One wave computes a 16 x N strip (N = 128 or 64).
// A 16x4 layout : lane = {row=lane&15}; v2f = A[row][k + 2*(lane>>4) + {0,1}]
//                 -> contiguous pair: single global_load_b64 per fragment.
// B 4x16 layout : lane = {col=lane&15}; v2f = B[k + 2*(lane>>4) + {0,1}][col]
// C 16x16 layout: acc[r] @ lane -> (M = r + 8*(lane>>4), Ncol = lane&15)
// Self and neighbor terms fused into one K loop: self then neighbor WMMA on the same acc
// per k-step across 8 accumulator chains.
// ---------------------------------------------------------------------------
template <int N, bool RELU>
__global__ void sage_gemm_kernel(const float* __restrict__ Hself,
                                 const float* __restrict__ Agg,
                                 const float* __restrict__ invdeg,
                                 const float* __restrict__ Ws,   // K x N
                                 const float* __restrict__ Wn,   // K x N
                                 const float* __restrict__ bias, // N
                                 float* __restrict__ Out,        // M x N
                                 int mtiles, int K) {
    const int lane  = threadIdx.x & 31;
    const int wave  = threadIdx.x >> 5;
    const int wpb   = blockDim.x >> 5;
    const int mtile = blockIdx.x * wpb + wave;
    if (mtile >= mtiles) return;  // wave-uniform: EXEC stays all-ones for WMMA

    const int row   = lane & 15;
    const int khalf = lane >> 4;        // 0: K k,k+1   1: K k+2,k+3
    const int gRow  = mtile * 16 + row;

    constexpr int NT = N / 16;
    v8f acc[NT];
#pragma unroll
    for (int t = 0; t < NT; ++t) acc[t] = v8f{};

    const float scale = invdeg[gRow];
    const float* hrow = Hself + (size_t)gRow * K;
    const float* arow = Agg   + (size_t)gRow * K;

    for (int k = 0; k < K; k += 4) {
        const int kb = k + khalf * 2;
        // A fragments: contiguous pairs -> b64 vector loads
        const v2f ah = *(const v2f*)(hrow + kb);
        v2f aa = *(const v2f*)(arow + kb);
        aa.x *= scale;
        aa.y *= scale;
#pragma unroll
        for (int t = 0; t < NT; ++t) {
            const int col = t * 16 + row;
            v2f bs, bn;
            bs.x = Ws[(size_t)kb * N + col];
            bs.y = Ws[(size_t)(kb + 1) * N + col];
            bn.x = Wn[(size_t)kb * N + col];
            bn.y = Wn[(size_t)(kb + 1) * N + col];
            acc[t] = __builtin_amdgcn_wmma_f32_16x16x4_f32(
                false, ah, false, bs, (short)0, acc[t], false, false);
            acc[t] = __builtin_amdgcn_wmma_f32_16x16x4_f32(
                false, aa, false, bn, (short)0, acc[t], false, false);
        }
    }

    // ---- epilogue: bias (+ ReLU) and store ----
    const int mbase = mtile * 16 + khalf * 8;
#pragma unroll
    for (int t = 0; t < NT; ++t) {
        const int col = t * 16 + row;
        const float bv = bias[col];
#pragma unroll
        for (int r = 0; r < 8; ++r) {
            float v = acc[t][r] + bv;
            if (RELU) v = fmaxf(v, 0.0f);
            Out[(size_t)(mbase + r) * N + col] = v;
        }
    }
}

// ---------------------------------------------------------------------------
// Host-side orchestration
// ---------------------------------------------------------------------------
extern "C" void kernel_launch(void* const* d_in, const int* in_sizes, int n_in,
                              void* d_out, int out_size, void* d_ws, size_t ws_size,
                              hipStream_t stream) {
    const float* x   = (const float*)d_in[0];
    const int*   src = (const int*)d_in[1];
    const int*   dst = (const int*)d_in[2];
    const float* ws0 = (const float*)d_in[3];
    const float* wn0 = (const float*)d_in[4];
    const float* b0  = (const float*)d_in[5];
    const float* ws1 = (const float*)d_in[6];
    const float* wn1 = (const float*)d_in[7];
    const float* b1  = (const float*)d_in[8];
    const float* ws2 = (const float*)d_in[9];
    const float* wn2 = (const float*)d_in[10];
    const float* b2  = (const float*)d_in[11];
    float* out = (float*)d_out;

    constexpr int IN = 128, HID = 128, OUT = 64;
    const int nNodes = in_sizes[0] / IN;
    const int nEdges = in_sizes[1];

    float* deg    = (float*)d_ws;
    float* invdeg = deg + nNodes;
    float* agg    = invdeg + nNodes;
    float* h1     = agg + (size_t)nNodes * HID;
    float* h2     = h1 + (size_t)nNodes * HID;

    // degrees
    hipMemsetAsync(deg, 0, sizeof(float) * nNodes, stream);
    sage_deg_kernel<<<(nEdges + 255) / 256, 256, 0, stream>>>(dst, deg, nEdges);
    sage_invdeg_kernel<<<(nNodes + 255) / 256, 256, 0, stream>>>(deg, invdeg, nNodes);

    const int mtiles = nNodes / 16;           // 100000 / 16 = 6250 exactly
    const int ggrd   = (mtiles + 3) / 4;      // 4 waves / block
    const dim3 gblk(128);

    const long long scatTot = (long long)nEdges * (HID / 4);
    const int scatGrd = (int)((scatTot + 255) / 256);

    // ---- layer 0: x -> h1 (ReLU) ----
    hipMemsetAsync(agg, 0, sizeof(float) * (size_t)nNodes * IN, stream);
    sage_scatter_kernel<128><<<scatGrd, 256, 0, stream>>>(x, src, dst, agg, nEdges);
    sage_gemm_kernel<128, true><<<ggrd, gblk, 0, stream>>>(
        x, agg, invdeg, ws0, wn0, b0, h1, mtiles, IN);

    // ---- layer 1: h1 -> h2 (ReLU) ----
    hipMemsetAsync(agg, 0, sizeof(float) * (size_t)nNodes * HID, stream);
    sage_scatter_kernel<128><<<scatGrd, 256, 0, stream>>>(h1, src, dst, agg, nEdges);
    sage_gemm_kernel<128, true><<<ggrd, gblk, 0, stream>>>(
        h1, agg, invdeg, ws1, wn1, b1, h2, mtiles, HID);

    // ---- layer 2: h2 -> out (no ReLU, N=64) ----
    hipMemsetAsync(agg, 0, sizeof(float) * (size_t)nNodes * HID, stream);
    sage_scatter_kernel<128><<<scatGrd, 256, 0, stream>>>(h2, src, dst, agg, nEdges);
    sage_gemm_kernel<64, false><<<ggrd, gblk, 0, stream>>>(
        h2, agg, invdeg, ws2, wn2, b2, out, mtiles, HID);
}